// DeepWave_57045755625642
// MI455X (gfx1250) — compile-verified
//
#include <hip/hip_runtime.h>
#include <math.h>

// ---------------------------------------------------------------------------
// DeepWave on MI455X (gfx1250), wave32 + V_WMMA_F32_16X16X4_F32.
//
// Pipeline (3 kernels on one stream):
//  1) build_w2i : W2i(2048x256) from D  (elementwise, tiny)
//  2) backproj  : Ypad(8192x256) = Sraw_f32(8192x2048) @ W2i  - tau   [WMMA f32]
//  3) cheb      : all 5 DeepWave iterations x 17 Chebyshev steps per block,
//                 L resident in LDS (256KB) + 64KB x-staging, WMMA f32.
// ---------------------------------------------------------------------------

typedef float v2f __attribute__((ext_vector_type(2)));
typedef float v8f __attribute__((ext_vector_type(8)));

#define NSMP   8192
#define NPX    242
#define NPAD   256
#define NCH    32
#define KD     2048          // 2 * 32 * 32 interleaved re/im
#define KCHEB  18
#define NITER  5
#define BETA   1.3130352854993312f   // 1/tanh(1)

__device__ __forceinline__ float retanh_f(float v) {
    float r = BETA * tanhf(v);
    return r > 0.0f ? r : 0.0f;
}

// ---------------------------------------------------------------------------
// Kernel 1: W2i[2q+0][p] = Dr_c*Dr_c' + Di_c*Di_c'   (coeff of Re S[q])
//           W2i[2q+1][p] = Di_c*Dr_c' - Dr_c*Di_c'   (coeff of Im S[q])
//           q = c*32+c', zero for p >= 242.
// ---------------------------------------------------------------------------
__global__ void build_w2i(const float* __restrict__ Dv, float* __restrict__ W2i) {
    int idx = blockIdx.x * blockDim.x + threadIdx.x;
    if (idx >= KD * NPAD) return;
    int k = idx >> 8;        // 0..2047
    int p = idx & 255;
    int q = k >> 1;
    int c  = q >> 5;
    int cc = q & 31;
    float v = 0.0f;
    if (p < NPX) {
        float drc  = Dv[(c  * NPX + p) * 2 + 0];
        float dic  = Dv[(c  * NPX + p) * 2 + 1];
        float drcc = Dv[(cc * NPX + p) * 2 + 0];
        float dicc = Dv[(cc * NPX + p) * 2 + 1];
        v = (k & 1) ? (dic * drcc - drc * dicc)
                    : (drc * drcc + dic * dicc);
    }
    W2i[idx] = v;
}

// ---------------------------------------------------------------------------
// Kernel 2: Ypad = Sraw @ W2i - tau.  4 waves/block, 16 rows/wave, N=256 wide.
// WMMA f32 16x16x4 layouts:
//   A 16x4 : lanes 0-15 row M=lane (K=k0,k0+1), lanes 16-31 row M=lane-16 (K=k0+2,k0+3)
//   B 4x16 : vgpr0 rows {k0, k0+2}, vgpr1 rows {k0+1, k0+3}; col = lane&15
//   C 16x16: vgpr r -> rows {r, r+8}; col = lane&15
// ---------------------------------------------------------------------------
__global__ void __launch_bounds__(128, 1)
backproj_kernel(const float* __restrict__ Sf, const float* __restrict__ W2i,
                const float* __restrict__ tau, float* __restrict__ ypad) {
    const int lane = threadIdx.x & 31;
    const int wave = threadIdx.x >> 5;
    const int row0 = blockIdx.x * 64 + wave * 16;
    const int lc   = lane & 15;
    const int sel  = (lane >> 4) << 1;     // 0 or 2 (K half-select)
    const int hi   = (lane >> 4) << 3;     // 0 or 8 (M half for C layout)

    v8f acc[16];
#pragma unroll
    for (int j = 0; j < 16; ++j) acc[j] = (v8f){0,0,0,0,0,0,0,0};

    const float* arow = Sf + (size_t)(row0 + lc) * KD;
    for (int k0 = 0; k0 < KD; k0 += 4) {
        v2f a;
        a.x = arow[k0 + sel + 0];
        a.y = arow[k0 + sel + 1];
#pragma unroll
        for (int j = 0; j < 16; ++j) {
            const float* bp = W2i + (size_t)(k0 + sel) * NPAD + j * 16 + lc;
            v2f b;
            b.x = bp[0];
            b.y = bp[NPAD];
            acc[j] = __builtin_amdgcn_wmma_f32_16x16x4_f32(
                false, a, false, b, (short)0, acc[j], false, false);
        }
    }
    // epilogue: subtract tau, write padded y (pad cols are exact zeros)
#pragma unroll
    for (int j = 0; j < 16; ++j) {
        int col = j * 16 + lc;
        float tv = (col < NPX) ? tau[col] : 0.0f;
#pragma unroll
        for (int r = 0; r < 8; ++r)
            ypad[(size_t)(row0 + r + hi) * NPAD + col] = acc[j][r] - tv;
    }
}

// ---------------------------------------------------------------------------
// Kernel 3: Chebyshev / DeepWave iterations.
// LDS: Lsh[256*256] (zero-padded L, row-major, stride 256) = 256 KB
//      xT [4 waves * 256 * 16] col-major staging per wave      =  64 KB
// Per-wave strip = 16 rows x 256 cols held as 16 v8f accumulators.
// ---------------------------------------------------------------------------
__device__ __forceinline__ void strip_storeT(float* __restrict__ xT, int lane,
                                             const v8f t[16], float scale) {
    const int hi = (lane >> 4) << 3;
    const int lc = lane & 15;
#pragma unroll
    for (int j = 0; j < 16; ++j) {
#pragma unroll
        for (int r = 0; r < 8; ++r)
            xT[(j * 16 + lc) * 16 + (r + hi)] = scale * t[j][r];
    }
}

__device__ __forceinline__ void strip_matmul(const float* __restrict__ Lsh,
                                             const float* __restrict__ xT,
                                             int lane, v8f dst[16]) {
    const int sel = (lane >> 4) << 1;
    const int lc  = lane & 15;
    for (int k0 = 0; k0 < NPAD; k0 += 4) {
        v2f a;
        a.x = xT[(k0 + sel + 0) * 16 + lc];
        a.y = xT[(k0 + sel + 1) * 16 + lc];
#pragma unroll
        for (int j = 0; j < 16; ++j) {
            const float* bp = Lsh + (size_t)(k0 + sel) * NPAD + j * 16 + lc;
            v2f b;
            b.x = bp[0];
            b.y = bp[NPAD];
            dst[j] = __builtin_amdgcn_wmma_f32_16x16x4_f32(
                false, a, false, b, (short)0, dst[j], false, false);
        }
    }
}

__global__ void __launch_bounds__(128, 1)
cheb_kernel(const float* __restrict__ Lg, const float* __restrict__ w,
            const float* __restrict__ ypad, float* __restrict__ out) {
    extern __shared__ float smem[];
    float* Lsh  = smem;                  // 65536 floats
    float* xbuf = smem + NPAD * NPAD;    // 16384 floats

    const int tid  = threadIdx.x;
    const int lane = tid & 31;
    const int wave = tid >> 5;

    // zero-pad + load L into LDS (242x242 -> 256x256)
    for (int i = tid; i < NPAD * NPAD; i += 128) Lsh[i] = 0.0f;
    __syncthreads();
    for (int i = tid; i < NPX * NPX; i += 128) {
        int r = i / NPX, c = i - r * NPX;
        Lsh[r * NPAD + c] = Lg[i];
    }
    __syncthreads();

    const int row0 = blockIdx.x * 64 + wave * 16;
    float* xT = xbuf + wave * (NPAD * 16);          // this wave's private staging
    const float* ybase = ypad + (size_t)row0 * NPAD;
    const int hi = (lane >> 4) << 3;
    const int lc = lane & 15;

    v8f A_[16], B_[16], ACC[16];

    // x = retanh(y)
#pragma unroll
    for (int j = 0; j < 16; ++j)
#pragma unroll
        for (int r = 0; r < 8; ++r)
            A_[j][r] = retanh_f(ybase[(size_t)(r + hi) * NPAD + j * 16 + lc]);

    for (int it = 0; it < NITER; ++it) {
        // T0 = x (in A_);  T1 = T0 @ L
        strip_storeT(xT, lane, A_, 1.0f);
#pragma unroll
        for (int j = 0; j < 16; ++j) B_[j] = (v8f){0,0,0,0,0,0,0,0};
        strip_matmul(Lsh, xT, lane, B_);

        float w0 = w[0], w1 = w[1];
#pragma unroll
        for (int j = 0; j < 16; ++j)
#pragma unroll
            for (int r = 0; r < 8; ++r)
                ACC[j][r] = w0 * A_[j][r] + w1 * B_[j][r];

        // T_{k+1} = (2*T_k) @ L - T_{k-1}, double-buffered in A_/B_
        for (int kk = 0; kk < (KCHEB - 2) / 2; ++kk) {
            // even step: new term -> A_
            strip_storeT(xT, lane, B_, 2.0f);
#pragma unroll
            for (int j = 0; j < 16; ++j)
#pragma unroll
                for (int r = 0; r < 8; ++r) A_[j][r] = -A_[j][r];
            strip_matmul(Lsh, xT, lane, A_);
            float wa = w[2 + 2 * kk];
#pragma unroll
            for (int j = 0; j < 16; ++j)
#pragma unroll
                for (int r = 0; r < 8; ++r) ACC[j][r] += wa * A_[j][r];

            // odd step: new term -> B_
            strip_storeT(xT, lane, A_, 2.0f);
#pragma unroll
            for (int j = 0; j < 16; ++j)
#pragma unroll
                for (int r = 0; r < 8; ++r) B_[j][r] = -B_[j][r];
            strip_matmul(Lsh, xT, lane, B_);
            float wb = w[3 + 2 * kk];
#pragma unroll
            for (int j = 0; j < 16; ++j)
#pragma unroll
                for (int r = 0; r < 8; ++r) ACC[j][r] += wb * B_[j][r];
        }

        // x = retanh(acc + y)
#pragma unroll
        for (int j = 0; j < 16; ++j)
#pragma unroll
            for (int r = 0; r < 8; ++r)
                A_[j][r] = retanh_f(ACC[j][r] +
                                    ybase[(size_t)(r + hi) * NPAD + j * 16 + lc]);
    }

    // write final x (only the 242 valid columns)
#pragma unroll
    for (int j = 0; j < 16; ++j) {
        int col = j * 16 + lc;
        if (col < NPX) {
#pragma unroll
            for (int r = 0; r < 8; ++r)
                out[(size_t)(row0 + r + hi) * NPX + col] = A_[j][r];
        }
    }
}

// ---------------------------------------------------------------------------
extern "C" void kernel_launch(void* const* d_in, const int* in_sizes, int n_in,
                              void* d_out, int out_size, void* d_ws, size_t ws_size,
                              hipStream_t stream) {
    const float* S   = (const float*)d_in[0];   // (8192,32,32) complex64 -> f32 x 2048/row
    const float* Dv  = (const float*)d_in[1];   // (32,242) complex64 interleaved
    const float* tau = (const float*)d_in[2];   // (242) f32
    const float* w   = (const float*)d_in[3];   // (18) f32
    const float* L   = (const float*)d_in[4];   // (242,242) f32
    float* out = (float*)d_out;                 // (8192,242) f32

    float* W2i  = (float*)d_ws;                 // 2048*256 f32 = 2 MB
    float* ypad = W2i + (size_t)KD * NPAD;      // 8192*256 f32 = 8 MB

    build_w2i<<<dim3((KD * NPAD + 255) / 256), dim3(256), 0, stream>>>(Dv, W2i);
    backproj_kernel<<<dim3(NSMP / 64), dim3(128), 0, stream>>>(S, W2i, tau, ypad);

    const size_t lds = (size_t)(NPAD * NPAD + 4 * NPAD * 16) * sizeof(float); // 320 KB
    cheb_kernel<<<dim3(NSMP / 64), dim3(128), lds, stream>>>(L, w, ypad, out);
}